// NeuralMemory_13769665151001
// MI455X (gfx1250) — compile-verified
//
#include <hip/hip_runtime.h>
#include <hip/hip_bf16.h>
#include <math.h>

// ---------------------------------------------------------------------------
// NeuralMemory (Titans-style fast-weight scan) for MI455X / gfx1250, wave32.
// B=4, T=64, H=256, D=1024.
//
// Phase 0: zero-pad W_fused (771x256) -> (784x256) so the WMMA GEMM loop is
//          branch-free (no EXEC save/restore around the B-operand loads).
// Phase 1: fused projection GEMM (256x771x256) via V_WMMA_F32_16X16X4_F32.
// Phase 2: activations (SiLU, L2-norm, sigmoid gates).
// Phase 3: state init: A0=W0+P0, A1=W1+P1 (W/P pairs provably receive
//          identical gradients -> fold into one matrix + shared momentum).
// Phase 4: sequential scan, one workgroup per batch; state is L2-resident.
// ---------------------------------------------------------------------------

typedef __attribute__((ext_vector_type(2))) float v2f;
typedef __attribute__((ext_vector_type(8))) float v8f;

#define Bn 4
#define Tn 64
#define Hn 256
#define Dn 1024
#define NF (3*Hn + 3)            // 771 fused-projection outputs
#define NFP 784                  // NF padded up to multiple of 16
#define MAT (Hn*Dn)              // 262144 elements per state matrix
#define STATE_STRIDE (4*MAT)     // A0,M0,A1,M1 per batch

__device__ __forceinline__ float sigf(float x) { return 1.0f / (1.0f + __expf(-x)); }

// ---------------------------------------------------------------------------
// Phase 0: Wfp[r][k] = (r < NF) ? Wf[r][k] : 0    (784 x 256)
// ---------------------------------------------------------------------------
__global__ void pad_wf(const float* __restrict__ Wf, float* __restrict__ Wfp) {
    const int r = blockIdx.x;
    const int k = threadIdx.x;
    Wfp[r * Hn + k] = (r < NF) ? Wf[r * Hn + k] : 0.0f;
}

// ---------------------------------------------------------------------------
// Phase 1: FP[m][n] = sum_k X[m][k] * Wfp[n][k]   (m<256, n<784, k<256)
// One wave (32 threads) per 16x16 output tile; K-loop in steps of 4.
// Fragment layouts per CDNA5 ISA 7.12.2 (fp32, wave32):
//   A (16x4): row = m0+(lane&15); VGPR0 holds K=2*(lane>>4), VGPR1 holds K+1
//   B (4x16): col = n0+(lane&15); same K striping as A
//   C/D     : vgpr v -> (M = m0 + v + 8*(lane>>4), N = n0 + (lane&15))
// Inner loop is branch-free: 2x global_load_b64 + 1x v_wmma.
// ---------------------------------------------------------------------------
__global__ void fused_proj_wmma(const float* __restrict__ X,
                                const float* __restrict__ Wfp,
                                float* __restrict__ FP) {
    const int lane = threadIdx.x;          // 0..31
    const int half = lane >> 4;            // 0/1
    const int l15  = lane & 15;
    const int n0   = blockIdx.x * 16;
    const int m0   = blockIdx.y * 16;
    const int row  = m0 + l15;             // A-matrix row this lane holds
    const int col  = n0 + l15;             // B-matrix col this lane holds

    const float* __restrict__ ap = X   + row * Hn + 2 * half;
    const float* __restrict__ bp = Wfp + col * Hn + 2 * half;

    v8f acc = {};
    #pragma unroll 8
    for (int kk = 0; kk < Hn; kk += 4) {
        const v2f a = *(const v2f*)(ap + kk);
        const v2f b = *(const v2f*)(bp + kk);
        acc = __builtin_amdgcn_wmma_f32_16x16x4_f32(
            /*neg_a=*/false, a, /*neg_b=*/false, b,
            /*c_mod=*/(short)0, acc, /*reuse_a=*/false, /*reuse_b=*/false);
    }
    if (col < NF) {
        #pragma unroll
        for (int v = 0; v < 8; ++v) {
            const int m = m0 + v + 8 * half;
            FP[m * NF + col] = acc[v];
        }
    }
}

// ---------------------------------------------------------------------------
// Phase 2: q = l2norm(silu(fp[0:H])), k = l2norm(silu(fp[H:2H])),
//          v = silu(fp[2H:3H]), alpha/theta/eta = sigmoid gates.
// One block of 256 threads per (b,t) row.
// ---------------------------------------------------------------------------
__global__ void act_kernel(const float* __restrict__ FP,
                           float* __restrict__ qb, float* __restrict__ kb,
                           float* __restrict__ vb, float* __restrict__ alp,
                           float* __restrict__ th, float* __restrict__ et) {
    __shared__ float red[Hn];
    const int bt = blockIdx.x;
    const int tid = threadIdx.x;
    const float* row = FP + bt * NF;

    const float qr = row[tid];
    const float kr = row[Hn + tid];
    const float vr = row[2 * Hn + tid];
    const float qsil = qr * sigf(qr);
    const float ksil = kr * sigf(kr);
    const float vsil = vr * sigf(vr);

    // ||silu(q)||
    red[tid] = qsil * qsil;
    __syncthreads();
    for (int s = 128; s > 0; s >>= 1) { if (tid < s) red[tid] += red[tid + s]; __syncthreads(); }
    const float qn = sqrtf(red[0]);
    __syncthreads();
    // ||silu(k)||
    red[tid] = ksil * ksil;
    __syncthreads();
    for (int s = 128; s > 0; s >>= 1) { if (tid < s) red[tid] += red[tid + s]; __syncthreads(); }
    const float kn = sqrtf(red[0]);

    qb[bt * Hn + tid] = qsil / fmaxf(qn, 1e-12f);
    kb[bt * Hn + tid] = ksil / fmaxf(kn, 1e-12f);
    vb[bt * Hn + tid] = vsil;
    if (tid == 0) {
        alp[bt] = sigf(row[3 * Hn]);
        th[bt]  = sigf(row[3 * Hn + 1]) * 0.1f;   // MAX_ADAPTIVE_LR
        et[bt]  = sigf(row[3 * Hn + 2]);
    }
}

// ---------------------------------------------------------------------------
// Phase 3: per-batch state init. A0 = W0+P0, A1 = W1+P1, M0 = M1 = 0.
// ---------------------------------------------------------------------------
__global__ void init_state(const float* __restrict__ W0, const float* __restrict__ P0,
                           const float* __restrict__ W1, const float* __restrict__ P1,
                           float* __restrict__ state) {
    const int e = blockIdx.x * blockDim.x + threadIdx.x;
    if (e >= MAT) return;
    const float a0 = W0[e] + P0[e];
    const float a1 = W1[e] + P1[e];
    #pragma unroll
    for (int b = 0; b < Bn; ++b) {
        float* st = state + (size_t)b * STATE_STRIDE;
        st[e]            = a0;    // A0 (H x D)
        st[MAT + e]      = 0.0f;  // M0
        st[2 * MAT + e]  = a1;    // A1 (D x H)
        st[3 * MAT + e]  = 0.0f;  // M1
    }
}

// ---------------------------------------------------------------------------
// Phase 4: sequential scan. One block (1024 threads = 32 waves) per batch.
// All matrix walks are lane-coalesced; A1^T * dpred uses wave shuffles.
// ---------------------------------------------------------------------------
__global__ __launch_bounds__(1024) void scan_kernel(
    const float* __restrict__ qb, const float* __restrict__ kb,
    const float* __restrict__ vb, const float* __restrict__ alp,
    const float* __restrict__ th, const float* __restrict__ et,
    float* __restrict__ state, float* __restrict__ out) {

    const int b = blockIdx.x;
    const int tid = threadIdx.x;            // 0..1023 : owns column c = tid of A0
    float* A0 = state + (size_t)b * STATE_STRIDE;
    float* M0 = A0 + MAT;
    float* A1 = M0 + MAT;                   // D x H, row-major
    float* M1 = A1 + MAT;

    __shared__ float ks[Hn], qs[Hn], vs[Hn];
    __shared__ float hs[Dn], gs[Dn], dhs[Dn], part[Dn];
    __shared__ float dps[Hn], red[Hn];
    __shared__ float s_alpha, s_theta, s_eta, s_qk;

    const int cg = tid >> 8;                // 0..3  (c-quarter for layer-2 passes)
    const int j  = tid & 255;               // output column for layer-2 passes
    const int wv = tid >> 5, ln = tid & 31; // wave / lane

    for (int t = 0; t < Tn; ++t) {
        const int bt = b * Tn + t;
        if (tid < Hn) {
            ks[tid] = kb[bt * Hn + tid];
            qs[tid] = qb[bt * Hn + tid];
            vs[tid] = vb[bt * Hn + tid];
        }
        if (tid == 0) { s_alpha = alp[bt]; s_theta = th[bt]; s_eta = et[bt]; }
        __syncthreads();

        // --- q . k (needed for q@M0' shortcut) ---
        if (tid < Hn) red[tid] = qs[tid] * ks[tid];
        __syncthreads();
        for (int s = 128; s > 0; s >>= 1) { if (tid < s) red[tid] += red[tid + s]; __syncthreads(); }
        if (tid == 0) s_qk = red[0];

        // --- Pass 1: hk = k@A0, hq = q@A0, qm = q@M0 (column walk, coalesced) ---
        float hk = 0.0f, hq = 0.0f, qm = 0.0f;
        {
            const float* a0c = A0 + tid;
            const float* m0c = M0 + tid;
            for (int r = 0; r < Hn; ++r) {
                __builtin_prefetch(a0c + (r + 8) * Dn, 0, 1);
                const float a = a0c[r * Dn];
                const float m = m0c[r * Dn];
                hk = fmaf(ks[r], a, hk);
                hq = fmaf(qs[r], a, hq);
                qm = fmaf(qs[r], m, qm);
            }
        }
        const float sg = sigf(hk);
        const float h = hk * sg;            // silu(k@A0)
        hs[tid] = h;
        __syncthreads();                    // hs + s_qk published

        // --- Pass 2: pred = h@A1 (4 partial threads per output j, coalesced) ---
        {
            float pp = 0.0f;
            const float* a1 = A1 + (size_t)(cg * 256) * Hn + j;
            const float* hp = hs + cg * 256;
            for (int i = 0; i < 256; ++i) pp = fmaf(hp[i], a1[(size_t)i * Hn], pp);
            part[tid] = pp;
        }
        __syncthreads();
        if (tid < Hn) {
            const float pr = part[tid] + part[tid + 256] + part[tid + 512] + part[tid + 768];
            dps[tid] = (2.0f / (float)Hn) * (pr - vs[tid]);  // dL/dpred
        }
        __syncthreads();

        // --- Pass 3: dh = dpred @ A1^T (each wave handles its own 32 rows) ---
        for (int i = 0; i < 32; ++i) {
            const int c = wv * 32 + i;
            const float* a1r = A1 + (size_t)c * Hn;
            float s = 0.0f;
            for (int jj = ln; jj < Hn; jj += 32) s = fmaf(a1r[jj], dps[jj], s);
            #pragma unroll
            for (int off = 16; off > 0; off >>= 1) s += __shfl_xor(s, off, 32);
            if (ln == 0) dhs[c] = s;
        }
        __syncthreads();

        // --- per-column: dh_pre, and updated-layer-1 output pre-activation ---
        const float dsil = sg * (1.0f + hk * (1.0f - sg));   // silu'(hk_pre)
        const float dh_pre = dhs[tid] * dsil;
        const float alpha = s_alpha, theta = s_theta, eta = s_eta, sqk = s_qk;
        // q@A0' = (1-a)(q@A0) + 2*(eta*(q@M0) - theta*(q.k)*dh_pre)
        const float hq1 = (1.0f - alpha) * hq + 2.0f * (eta * qm - theta * sqk * dh_pre);
        const float g = hq1 * sigf(hq1);     // silu(q@A0')
        gs[tid] = g;
        __syncthreads();                     // gs published; old-A1 reads done

        // --- Pass 4: rank-1 update of M0/A0 (column walk, coalesced) ---
        {
            float* m0c = M0 + tid;
            float* a0c = A0 + tid;
            for (int r = 0; r < Hn; ++r) {
                const int idx = r * Dn;
                const float m = eta * m0c[idx] - theta * ks[r] * dh_pre;
                m0c[idx] = m;
                a0c[idx] = (1.0f - alpha) * a0c[idx] + 2.0f * m;
            }
        }

        // --- Pass 5: rank-1 update of M1/A1 fused with out = silu(q@A0') @ A1' ---
        {
            const float dj = dps[j];
            float op = 0.0f;
            float* a1 = A1 + (size_t)(cg * 256) * Hn + j;
            float* m1 = M1 + (size_t)(cg * 256) * Hn + j;
            const float* hp = hs + cg * 256;
            const float* gp = gs + cg * 256;
            for (int i = 0; i < 256; ++i) {
                const size_t idx = (size_t)i * Hn;
                const float m = eta * m1[idx] - theta * hp[i] * dj;
                m1[idx] = m;
                const float a = (1.0f - alpha) * a1[idx] + 2.0f * m;
                a1[idx] = a;
                op = fmaf(gp[i], a, op);
            }
            part[tid] = op;
        }
        __syncthreads();
        if (tid < Hn) {
            out[bt * Hn + tid] = part[tid] + part[tid + 256] + part[tid + 512] + part[tid + 768];
        }
        __syncthreads();   // protect shared buffers before next step
    }
}

// ---------------------------------------------------------------------------
// Host-side launcher.
// Inputs (setup_inputs order): x, W_fused, W0, W1, P0, P1  (all fp32)
// Output: (B,T,H) fp32 = 65536 floats.
// ---------------------------------------------------------------------------
extern "C" void kernel_launch(void* const* d_in, const int* in_sizes, int n_in,
                              void* d_out, int out_size, void* d_ws, size_t ws_size,
                              hipStream_t stream) {
    const float* x  = (const float*)d_in[0];
    const float* Wf = (const float*)d_in[1];
    const float* W0 = (const float*)d_in[2];
    const float* W1 = (const float*)d_in[3];
    const float* P0 = (const float*)d_in[4];
    const float* P1 = (const float*)d_in[5];
    float* outp = (float*)d_out;
    float* ws = (float*)d_ws;

    // Workspace layout (floats)
    const size_t off_fp = 0;                               // B*T x 771
    const size_t off_wp = off_fp + (size_t)Bn * Tn * NF;   // padded Wf: 784x256
    const size_t off_q  = off_wp + (size_t)NFP * Hn;
    const size_t off_k  = off_q + (size_t)Bn * Tn * Hn;
    const size_t off_v  = off_k + (size_t)Bn * Tn * Hn;
    const size_t off_al = off_v + (size_t)Bn * Tn * Hn;
    const size_t off_th = off_al + (size_t)Bn * Tn;
    const size_t off_et = off_th + (size_t)Bn * Tn;
    const size_t off_st = off_et + (size_t)Bn * Tn;        // 4 * (4*MAT) floats

    float* fp  = ws + off_fp;
    float* wfp = ws + off_wp;
    float* qb  = ws + off_q;
    float* kb  = ws + off_k;
    float* vb  = ws + off_v;
    float* alp = ws + off_al;
    float* thb = ws + off_th;
    float* etb = ws + off_et;
    float* st  = ws + off_st;

    // Phase 0: zero-pad W_fused so the WMMA loop needs no EXEC branching
    pad_wf<<<NFP, Hn, 0, stream>>>(Wf, wfp);
    // Phase 1: projection GEMM via WMMA (16 M-tiles x 49 N-tiles, 1 wave each)
    fused_proj_wmma<<<dim3(NFP / 16, (Bn * Tn) / 16), 32, 0, stream>>>(x, wfp, fp);
    // Phase 2: activations
    act_kernel<<<Bn * Tn, Hn, 0, stream>>>(fp, qb, kb, vb, alp, thb, etb);
    // Phase 3: state init
    init_state<<<(MAT + 255) / 256, 256, 0, stream>>>(W0, P0, W1, P1, st);
    // Phase 4: sequential fast-weight scan
    scan_kernel<<<Bn, 1024, 0, stream>>>(qb, kb, vb, alp, thb, etb, st, outp);
}